// Attention_66245575573729
// MI455X (gfx1250) — compile-verified
//
#include <hip/hip_runtime.h>
#include <hip/hip_bf16.h>

typedef float v2f __attribute__((ext_vector_type(2)));
typedef float v8f __attribute__((ext_vector_type(8)));

#define HW     16384
#define CIN    192
#define WIMG   128
#define NCHUNK 12            // 192 / 16 K-chunks
#define PSTR   288           // dwords per LDS pair-row (256 data + 32 pad; 288%64==32)
#define BUFDW  (8 * PSTR)    // dwords per LDS buffer (8 row-pairs)

// ---------------------------------------------------------------------------
// Issue one 16x128 B tile (rows k0..k0+15, cols nBlock..+127) as async
// global->LDS copies. Each thread moves 8 dwords: col c = tid&127,
// parity par = tid>>7, rows r = 2i+par. LDS layout is pair-interleaved:
// element (r,c) at dword (r>>1)*PSTR + 2c + (r&1), so each WMMA B fragment
// (two consecutive K rows, same col) is one aligned ds_load_b64.
// ---------------------------------------------------------------------------
__device__ __forceinline__
void issue_tile_async(unsigned long long gBase, int k0, int nBlock,
                      unsigned int ldsBufByte, int tid)
{
    const int c   = tid & 127;
    const int par = tid >> 7;
    unsigned int lofs = ldsBufByte + (unsigned int)((c * 2 + par) * 4);
    unsigned int gofs = (unsigned int)((((k0 + par) * HW) + nBlock + c) * 4);
#pragma unroll
    for (int i = 0; i < 8; ++i) {
        asm volatile("global_load_async_to_lds_b32 %0, %1, %2"
                     :: "v"(lofs), "v"(gofs), "s"(gBase)
                     : "memory");
        lofs += PSTR * 4;
        gofs += 2 * HW * 4;
    }
}

// ---------------------------------------------------------------------------
// GEMM: C[b] (M x HW) = A[b] (M x 192, row-major) @ B[b] (192 x HW, row-major)
// Block = 256 threads (8 waves), tile 64(M) x 128(N); wave strip 16 x 64 ->
// 4 f32 WMMA accumulators. B tiles double-buffered in LDS via async DMA.
// ---------------------------------------------------------------------------
__global__ __launch_bounds__(256)
void gemm_wmma_f32(const float* __restrict__ A, long aBatchStride,
                   const float* __restrict__ B, long bBatchStride,
                   float* __restrict__ C, long cBatchStride)
{
    const int b      = blockIdx.z;
    const int mBlock = blockIdx.y * 64;
    const int nBlock = blockIdx.x * 128;
    const int tid    = threadIdx.x;
    const int lane   = tid & 31;
    const int wave   = tid >> 5;
    const int wm     = wave & 3;   // M sub-tile (16 rows)
    const int wn     = wave >> 2;  // N sub-tile (64 cols)

    const float* Ab = A + (long)b * aBatchStride;
    const float* Bb = B + (long)b * bBatchStride;
    float*       Cb = C + (long)b * cBatchStride;

    __shared__ __align__(16) float ldsB[2 * BUFDW];
    const unsigned int       ldsBase = (unsigned int)(uintptr_t)(void*)ldsB;
    const unsigned long long gBase   = (unsigned long long)(uintptr_t)Bb;

    const int m    = mBlock + wm * 16 + (lane & 15);
    const int half = lane >> 4;
    const int koff = half * 2;                 // half-wave selects K pair

    v8f acc[4];
#pragma unroll
    for (int j = 0; j < 4; ++j)
#pragma unroll
        for (int r = 0; r < 8; ++r) acc[j][r] = 0.0f;

    // prologue: start tile 0
    issue_tile_async(gBase, 0, nBlock, ldsBase, tid);

    for (int c = 0; c < NCHUNK; ++c) {
        if (c + 1 < NCHUNK) {
            // start next tile into the other buffer, then retire only tile c
            issue_tile_async(gBase, (c + 1) * 16, nBlock,
                             ldsBase + ((c + 1) & 1) * (BUFDW * 4), tid);
            asm volatile("s_wait_asynccnt 0x8" ::: "memory");
        } else {
            asm volatile("s_wait_asynccnt 0x0" ::: "memory");
        }
        __syncthreads();   // tile c visible to all waves

        const float* bufp = ldsB + (c & 1) * BUFDW;
        const int    k0   = c * 16;
#pragma unroll
        for (int kk = 0; kk < 16; kk += 4) {
            // A fragment: 16x4 f32, lane=M, half-wave -> K {0,1} / {2,3}
            const float* ap = Ab + (long)m * CIN + (k0 + kk + koff);
            v2f a;
            a.x = ap[0];
            a.y = ap[1];
            const float* bp = bufp + ((kk >> 1) + half) * PSTR
                                   + (wn * 64 + (lane & 15)) * 2;
#pragma unroll
            for (int j = 0; j < 4; ++j) {
                const v2f bf = *(const v2f*)(bp + j * 32);   // one ds_load_b64
                acc[j] = __builtin_amdgcn_wmma_f32_16x16x4_f32(
                    false, a, false, bf, (short)0, acc[j], false, false);
            }
        }
        __syncthreads();   // done reading buffer (c&1) before it is refilled
    }

    // C/D layout: VGPR r, lanes 0-15 -> M=r ; lanes 16-31 -> M=r+8
    const int mBase = mBlock + wm * 16 + half * 8;
#pragma unroll
    for (int j = 0; j < 4; ++j) {
        const int ncol = nBlock + wn * 64 + j * 16 + (lane & 15);
#pragma unroll
        for (int r = 0; r < 8; ++r)
            Cb[(long)(mBase + r) * HW + ncol] = acc[j][r];
    }
}

// ---------------------------------------------------------------------------
// Depthwise 3x3, SAME zero padding. One block per (b, ch, row).
// ---------------------------------------------------------------------------
__global__ __launch_bounds__(128)
void dwconv3x3(const float* __restrict__ in, const float* __restrict__ w,
               float* __restrict__ out)
{
    const int y  = blockIdx.x;
    const int ch = blockIdx.y;
    const int b  = blockIdx.z;
    const int x  = threadIdx.x;

    const float* wp    = w + ch * 9;          // uniform per block -> SGPR loads
    const long   cbase = ((long)b * 576 + ch) * (long)HW;
    const float* cin   = in + cbase;

    float acc = 0.0f;
#pragma unroll
    for (int dy = -1; dy <= 1; ++dy) {
        const int yy = y + dy;
        if (yy < 0 || yy > WIMG - 1) continue;
        const float* rp = cin + yy * WIMG;
#pragma unroll
        for (int dx = -1; dx <= 1; ++dx) {
            const int xx = x + dx;
            if (xx < 0 || xx > WIMG - 1) continue;
            acc += wp[(dy + 1) * 3 + (dx + 1)] * rp[xx];
        }
    }
    out[cbase + y * WIMG + x] = acc;
}

// ---------------------------------------------------------------------------
// Zero-fill accumulation buffers (Gram + sumsq) before atomic accumulation.
// ---------------------------------------------------------------------------
__global__ __launch_bounds__(256)
void zero_f32(float* __restrict__ p, int n)
{
    int i = blockIdx.x * 256 + threadIdx.x;
    if (i < n) p[i] = 0.0f;
}

// ---------------------------------------------------------------------------
// Gram partials: per (b,head, s-chunk of 2048) compute partial
//   G[d,e]  += sum_s q[d,s]*k[e,s]     (48x48)
//   sq_q[d] += sum_s q[d,s]^2,  sq_k[e] += sum_s k[e,s]^2
// 256 threads = 16x16 grid of 3x3 register tiles over (d,e).
// ---------------------------------------------------------------------------
#define SCH 2048
#define SUB 64
__global__ __launch_bounds__(256)
void gram_partial(const float* __restrict__ qkv, float* __restrict__ G,
                  float* __restrict__ sq_q, float* __restrict__ sq_k)
{
    const int chunk = blockIdx.x;      // 0..7
    const int bh    = blockIdx.y;      // 0..31  (= b*4 + head)
    const int b     = bh >> 2;
    const int head  = bh & 3;
    const long qbase = ((long)b * 576 + head * 48) * (long)HW;
    const long kbase = ((long)b * 576 + 192 + head * 48) * (long)HW;
    const int  s0    = chunk * SCH;
    const int  tid   = threadIdx.x;

    __shared__ float lq[48][SUB + 1];
    __shared__ float lk[48][SUB + 1];

    const int td = (tid & 15) * 3;     // d tile base
    const int te = (tid >> 4) * 3;     // e tile base

    float g[3][3] = {};
    float sq[3]   = {};
    float sk[3]   = {};

    for (int ss = 0; ss < SCH; ss += SUB) {
        __syncthreads();
        for (int i = tid; i < 48 * SUB; i += 256) {
            const int r = i / SUB, c = i % SUB;
            lq[r][c] = qkv[qbase + (long)r * HW + s0 + ss + c];
            lk[r][c] = qkv[kbase + (long)r * HW + s0 + ss + c];
        }
        __syncthreads();
        for (int s = 0; s < SUB; ++s) {
            float qv[3], kv[3];
#pragma unroll
            for (int i = 0; i < 3; ++i) { qv[i] = lq[td + i][s]; kv[i] = lk[te + i][s]; }
#pragma unroll
            for (int i = 0; i < 3; ++i)
#pragma unroll
                for (int j = 0; j < 3; ++j) g[i][j] += qv[i] * kv[j];
            if ((tid >> 4) == 0) {
#pragma unroll
                for (int i = 0; i < 3; ++i) sq[i] += qv[i] * qv[i];
            }
            if ((tid >> 4) == 1) {
#pragma unroll
                for (int i = 0; i < 3; ++i) { float t = lk[td + i][s]; sk[i] += t * t; }
            }
        }
    }

    float* gp = G + (long)bh * 48 * 48;
#pragma unroll
    for (int i = 0; i < 3; ++i)
#pragma unroll
        for (int j = 0; j < 3; ++j)
            atomicAdd(&gp[(td + i) * 48 + (te + j)], g[i][j]);
    if ((tid >> 4) == 0)
#pragma unroll
        for (int i = 0; i < 3; ++i) atomicAdd(&sq_q[bh * 48 + td + i], sq[i]);
    if ((tid >> 4) == 1)
#pragma unroll
        for (int i = 0; i < 3; ++i) atomicAdd(&sq_k[bh * 48 + td + i], sk[i]);
}

// ---------------------------------------------------------------------------
// Per batch: normalize Gram -> softmax -> compose M[b] = proj_w . attn_blocks
//   attn[h,d,e] = softmax_e( G[h,d,e] / (max(|q_d|,eps)*max(|k_e|,eps)) * T[h] )
//   M[o, h*48+e] = sum_d proj_w[o, h*48+d] * attn[h,d,e]
// ---------------------------------------------------------------------------
__global__ __launch_bounds__(256)
void softmax_compose(const float* __restrict__ G, const float* __restrict__ sq_q,
                     const float* __restrict__ sq_k, const float* __restrict__ temp,
                     const float* __restrict__ projw, float* __restrict__ M)
{
    const int b   = blockIdx.x;
    const int tid = threadIdx.x;
    __shared__ float A[4][48][49];

    if (tid < 192) {
        const int head = tid / 48, d = tid % 48;
        const float* gr = G + (((long)b * 4 + head) * 48 + d) * 48;
        const float nq  = fmaxf(sqrtf(sq_q[(b * 4 + head) * 48 + d]), 1e-12f);
        const float t   = temp[head];
        float row[48];
        float mx = -3.402823466e+38f;
#pragma unroll
        for (int e = 0; e < 48; ++e) {
            const float nk = fmaxf(sqrtf(sq_k[(b * 4 + head) * 48 + e]), 1e-12f);
            const float v  = gr[e] / (nq * nk) * t;
            row[e] = v;
            mx = fmaxf(mx, v);
        }
        float sum = 0.0f;
#pragma unroll
        for (int e = 0; e < 48; ++e) { const float ex = __expf(row[e] - mx); row[e] = ex; sum += ex; }
        const float inv = 1.0f / sum;
#pragma unroll
        for (int e = 0; e < 48; ++e) A[head][d][e] = row[e] * inv;
    }
    __syncthreads();

    for (int idx = tid; idx < 192 * 192; idx += 256) {
        const int o = idx / 192, cc = idx % 192;
        const int head = cc / 48, e = cc % 48;
        const float* pw = projw + o * 192 + head * 48;
        float acc = 0.0f;
#pragma unroll
        for (int d = 0; d < 48; ++d) acc += pw[d] * A[head][d][e];
        M[((long)b * 192 + o) * 192 + cc] = acc;
    }
}

// ---------------------------------------------------------------------------
// Launch: x, qkv_w, dw_w, proj_w, temperature  ->  out [8,192,128,128] f32
// Workspace layout (floats):
//   qkv_lin  [8,576,16384]   75,497,472
//   qkv_dw   [8,576,16384]   75,497,472
//   G        [8,4,48,48]     73,728
//   sq_q     [8,4,48]        1,536
//   sq_k     [8,4,48]        1,536
//   M        [8,192,192]     294,912
// ---------------------------------------------------------------------------
extern "C" void kernel_launch(void* const* d_in, const int* in_sizes, int n_in,
                              void* d_out, int out_size, void* d_ws, size_t ws_size,
                              hipStream_t stream)
{
    (void)in_sizes; (void)n_in; (void)out_size; (void)ws_size;
    const float* x      = (const float*)d_in[0];
    const float* qkv_w  = (const float*)d_in[1];
    const float* dw_w   = (const float*)d_in[2];
    const float* proj_w = (const float*)d_in[3];
    const float* temp   = (const float*)d_in[4];
    float* out = (float*)d_out;
    float* ws  = (float*)d_ws;

    float* qkv_lin = ws;
    float* qkv_dw  = ws + 75497472L;
    float* G       = ws + 150994944L;
    float* sq_q    = G + 73728;
    float* sq_k    = sq_q + 1536;
    float* Mbuf    = sq_k + 1536;

    // K1: qkv_lin[b] = qkv_w (576x192) @ x[b] (192x16384)
    gemm_wmma_f32<<<dim3(128, 9, 8), 256, 0, stream>>>(
        qkv_w, 0L, x, (long)CIN * HW, qkv_lin, (long)576 * HW);

    // K2: depthwise 3x3
    dwconv3x3<<<dim3(128, 576, 8), 128, 0, stream>>>(qkv_lin, dw_w, qkv_dw);

    // K0: zero Gram + sumsq accumulators
    zero_f32<<<(76800 + 255) / 256, 256, 0, stream>>>(G, 76800);

    // K3: Gram partials + row norms
    gram_partial<<<dim3(8, 32), 256, 0, stream>>>(qkv_dw, G, sq_q, sq_k);

    // K4: softmax + compose M[b] = proj_w . attn
    softmax_compose<<<8, 256, 0, stream>>>(G, sq_q, sq_k, temp, proj_w, Mbuf);

    // K5: out[b] = M[b] (192x192) @ v[b] (192x16384), v = qkv_dw rows 384..575
    gemm_wmma_f32<<<dim3(128, 3, 8), 256, 0, stream>>>(
        Mbuf, (long)CIN * CIN, qkv_dw + (long)384 * HW, (long)576 * HW,
        out, (long)CIN * HW);
}